// DS3_Dempster_601295422077
// MI455X (gfx1250) — compile-verified
//
#include <hip/hip_runtime.h>
#include <stdint.h>

// -------- problem geometry --------
#define TSTEPS 64
#define VEC 16
#define ROWS_PER_BLOCK 64
#define CH 4                               // timesteps per staged chunk
#define NCHUNK (TSTEPS / CH)               // 16 chunks
#define ROW_BYTES_G (TSTEPS * VEC * 4)     // 4096 B per row in global
#define CHUNK_BYTES (CH * VEC * 4)         // 256 B per row per chunk
#define ROW_F_LDS 68                       // 64 data floats + 4 pad (bank decorrelation)
#define ROW_B_LDS (ROW_F_LDS * 4)          // 272 B
#define BUF_F (ROWS_PER_BLOCK * ROW_F_LDS) // floats per LDS buffer
#define EPSF 1e-7f

// Cooperative async fetch of one 64-row x CH-timestep chunk, GVS mode:
//   saddr  = block-uniform SGPR pair (in + row0*4096)
//   voff   = 32-bit per-thread offset: row (a+4j)*4096 + slice b*16 + chunk c*256
// Thread (a = tid>>4, b = tid&15) issues 16 x b128 async loads; consecutive
// tids cover consecutive 16B of the same 256B row-chunk, so every wave
// instruction touches whole cachelines. TH_LOAD_NT: 512MB one-pass stream
// (> 192MB L2) must not thrash the caches. Data lands in LDS via the async
// pipe (ASYNCcnt), never passing through VGPRs.
__device__ __forceinline__ void prefetch_chunk(unsigned long long sbase,
                                               unsigned voff, unsigned l) {
#pragma unroll
    for (int j = 0; j < 16; ++j) {
        asm volatile("global_load_async_to_lds_b128 %0, %1, %2 th:TH_LOAD_NT"
                     :
                     : "v"(l), "v"(voff), "s"(sbase)
                     : "memory");
        voff += 4u * ROW_BYTES_G;  // 4 rows ahead
        l += 4u * ROW_B_LDS;
    }
}

// One Dempster combination step: m <- normalize(m*v + m*v[15] + m[15]*v)
__device__ __forceinline__ void combine(float m[VEC], const float* __restrict__ v) {
    const float w1 = m[15];
    const float w2 = v[15];
    float c[VEC];
    float s = 0.0f;
#pragma unroll
    for (int k = 0; k < VEC; ++k) {
        c[k] = fmaf(w1, v[k], m[k] * (v[k] + w2));
        s += c[k];
    }
    const float r = 1.0f / (s + EPSF);
#pragma unroll
    for (int k = 0; k < VEC; ++k) m[k] = c[k] * r;
}

__global__ void __launch_bounds__(ROWS_PER_BLOCK)
ds3_dempster_scan(const float* __restrict__ in, float* __restrict__ out, int nrows) {
    __shared__ float lds[2][BUF_F];

    const int tid = threadIdx.x;
    const int a = tid >> 4;
    const int b = tid & 15;
    const unsigned long long row0 = (unsigned long long)blockIdx.x * ROWS_PER_BLOCK;

    // Block-uniform 64-bit base (stays in SGPRs) + small per-thread 32-bit offset.
    const unsigned long long sbase =
        (unsigned long long)(uintptr_t)in + row0 * (unsigned long long)ROW_BYTES_G;
    const unsigned voff0 = (unsigned)a * ROW_BYTES_G + (unsigned)b * 16u;

    // LDS byte offsets: flat shared pointer low 32 bits == LDS offset
    // (ISA aperture rule: the low 32 bits are the LDS address).
    const unsigned lds0 = (unsigned)(uintptr_t)(&lds[0][0]);
    const unsigned lds1 = (unsigned)(uintptr_t)(&lds[1][0]);
    const unsigned lane_l = (unsigned)a * ROW_B_LDS + (unsigned)b * 16u;

    // warm the pipeline: chunk 0 -> buffer 0
    prefetch_chunk(sbase, voff0, lds0 + lane_l);

    float m[VEC];

#pragma unroll 1
    for (int c = 0; c < NCHUNK; ++c) {
        if (c + 1 < NCHUNK) {
            // Issue next chunk into the other buffer, then drain down to it:
            // <=16 outstanding leaves exactly the just-issued prefetch in
            // flight (async loads complete in order), so chunk c is in LDS.
            prefetch_chunk(sbase, voff0 + (unsigned)(c + 1) * CHUNK_BYTES,
                           ((c & 1) ? lds0 : lds1) + lane_l);
            asm volatile("s_wait_asynccnt 16" ::: "memory");
        } else {
            asm volatile("s_wait_asynccnt 0" ::: "memory");
        }
        __syncthreads();  // make other waves' async writes visible

        const float* r = ((c & 1) ? &lds[1][0] : &lds[0][0]) + tid * ROW_F_LDS;
        if (c == 0) {
#pragma unroll
            for (int k = 0; k < VEC; ++k) m[k] = r[k];  // m0 = x[t=0]
            combine(m, r + 1 * VEC);
            combine(m, r + 2 * VEC);
            combine(m, r + 3 * VEC);
        } else {
#pragma unroll
            for (int tl = 0; tl < CH; ++tl) combine(m, r + tl * VEC);
        }
        __syncthreads();  // buffer we just read gets overwritten next iter
    }

    const unsigned long long row = row0 + (unsigned)tid;
    if (row < (unsigned long long)(unsigned)nrows) {
        float* o = out + row * (unsigned long long)VEC;
#pragma unroll
        for (int k = 0; k < VEC; ++k) o[k] = m[k];
    }
}

extern "C" void kernel_launch(void* const* d_in, const int* in_sizes, int n_in,
                              void* d_out, int out_size, void* d_ws, size_t ws_size,
                              hipStream_t stream) {
    (void)n_in; (void)out_size; (void)d_ws; (void)ws_size;
    const float* in = (const float*)d_in[0];
    float* out = (float*)d_out;

    const long long total = in_sizes[0];                 // 131072*64*16
    const int nrows = (int)(total / (TSTEPS * VEC));     // 131072
    const int blocks = nrows / ROWS_PER_BLOCK;           // 2048 (exact)

    ds3_dempster_scan<<<blocks, ROWS_PER_BLOCK, 0, stream>>>(in, out, nrows);
}